// FourierFourierAttention_71305047048413
// MI455X (gfx1250) — compile-verified
//
#include <hip/hip_runtime.h>

typedef float v2f __attribute__((ext_vector_type(2)));
typedef float v8f __attribute__((ext_vector_type(8)));

#define NB   16
#define NSEQ 8192
#define NSZ  128
#define LOG2_NSEQ 13

// ---------------------------------------------------------------------------
// Kernel 0: build symmetric DFT matrices in workspace.
//   fcos[i][j] =  cos(2*pi*i*j/128)
//   fsin[i][j] = -sin(2*pi*i*j/128)   (minus baked in: f32 WMMA has no A-neg)
// ---------------------------------------------------------------------------
__global__ __launch_bounds__(256) void k_init_dft(float* __restrict__ fcos,
                                                  float* __restrict__ fsin) {
    int idx = blockIdx.x * 256 + threadIdx.x;           // 0..16383
    int i = idx >> 7, j = idx & 127;
    int m = (i * j) & 127;                              // exact index math
    float ang = (float)m * (6.28318530717958647692f / 128.0f);
    fcos[idx] =  cosf(ang);
    fsin[idx] = -sinf(ang);
}

// ---------------------------------------------------------------------------
// Kernel 1: forward 128-pt DFT over size dim via f32 WMMA.
//   in : x[b][s][c]      (real)
//   out: wre/wim[b][k][s] (seq-contiguous for stage 2)
//   Re = X*C ; Im = X*(-S)  (fsin already holds -sin)
// One block = 16 rows (b,s). 8 waves, each owns one 16-wide k-tile.
// ---------------------------------------------------------------------------
__global__ __launch_bounds__(256)
void k_dft_size_fwd(const float* __restrict__ x,
                    const float* __restrict__ fcos,
                    const float* __restrict__ fsin,
                    float* __restrict__ wre, float* __restrict__ wim) {
    __shared__ float xs[16 * 130];       // padded: conflict-free A-fragment reads
    __shared__ float obR[128 * 17];      // [k][s] staging, padded
    __shared__ float obI[128 * 17];

    const int tid  = threadIdx.x;
    const int row0 = blockIdx.x * 16;    // global row = b*8192 + s
    const int b    = row0 >> 13;
    const int s0   = row0 & (NSEQ - 1);

    // load 16x128 input tile (coalesced), store padded
    for (int i = tid; i < 16 * 128; i += 256) {
        int m = i >> 7, c = i & 127;
        xs[m * 130 + c] = x[(size_t)(row0 + m) * NSZ + c];
    }
    __syncthreads();

    const int wave = tid >> 5, lane = tid & 31;
    const int hi = lane >> 4, lm = lane & 15;
    const int n = wave * 16 + lm;        // frequency column (B matrix col)

    v8f accR = {}; v8f accI = {};
    for (int k4 = 0; k4 < 128; k4 += 4) {
        const int kb = k4 + 2 * hi;      // A/B fragment K rows per ISA layout
        v2f a;  a.x  = xs[lm * 130 + kb];     a.y  = xs[lm * 130 + kb + 1];
        v2f bc; bc.x = fcos[n * 128 + kb];    bc.y = fcos[n * 128 + kb + 1]; // F symmetric
        v2f bs; bs.x = fsin[n * 128 + kb];    bs.y = fsin[n * 128 + kb + 1];
        accR = __builtin_amdgcn_wmma_f32_16x16x4_f32(false, a, false, bc, (short)0, accR, false, false);
        accI = __builtin_amdgcn_wmma_f32_16x16x4_f32(false, a, false, bs, (short)0, accI, false, false);
    }

    // D layout: vgpr r, lanes 0-15 -> M=r, lanes 16-31 -> M=r+8, N=lane&15
    for (int r = 0; r < 8; ++r) {
        int m = r + 8 * hi;              // local s
        int k = wave * 16 + lm;          // frequency
        obR[k * 17 + m] = accR[r];
        obI[k * 17 + m] = accI[r];
    }
    __syncthreads();

    // coalesced store: for each k, 16 consecutive s (64B segments)
    for (int i = tid; i < 128 * 16; i += 256) {
        int k = i >> 4, s = i & 15;
        size_t off = ((size_t)(b * NSZ + k)) * NSEQ + s0 + s;
        wre[off] = obR[k * 17 + s];
        wim[off] = obI[k * 17 + s];
    }
}

// ---------------------------------------------------------------------------
// Kernel 2: per-(b,kfreq) column: fft_seq -> pointwise -> ifft_seq -> *rw.
// 8192-pt radix-2 FFT entirely in 64KB dynamic LDS. Forward = DIT
// (bit-rev load), inverse = DIF (bit-rev store) -> no in-LDS permutation.
// ---------------------------------------------------------------------------
__global__ __launch_bounds__(256)
void k_seq_fft(float* __restrict__ wre, float* __restrict__ wim,
               const float* __restrict__ qr, const float* __restrict__ qi,
               const float* __restrict__ kr, const float* __restrict__ ki,
               const float* __restrict__ vr, const float* __restrict__ vi,
               const float* __restrict__ rr, const float* __restrict__ ri) {
    extern __shared__ float lds[];
    float* re = lds;
    float* im = lds + NSEQ;

    const int tid = threadIdx.x;
    const int col = blockIdx.x;            // b*128 + kfreq
    const int kf  = col & 127;
    const size_t base = (size_t)col * NSEQ;

    // bit-reversed load (DIT input ordering)
    for (int i = tid; i < NSEQ; i += 256) {
        int j = (int)(__brev((unsigned)i) >> (32 - LOG2_NSEQ));
        re[j] = wre[base + i];
        im[j] = wim[base + i];
    }
    __syncthreads();

    // forward FFT (sign = -1), DIT
    for (int len = 1; len < NSEQ; len <<= 1) {
        const float ang0 = -3.14159265358979323846f / (float)len;
        for (int t = tid; t < NSEQ / 2; t += 256) {
            int pos = t & (len - 1);
            int i0  = ((t & ~(len - 1)) << 1) | pos;
            int i1  = i0 + len;
            float c, s; __sincosf(ang0 * (float)pos, &s, &c);
            float ur = re[i0], ui = im[i0];
            float xr = re[i1], xi = im[i1];
            float tr = xr * c - xi * s;
            float ti = xr * s + xi * c;
            re[i0] = ur + tr; im[i0] = ui + ti;
            re[i1] = ur - tr; im[i1] = ui - ti;
        }
        __syncthreads();
    }

    // pointwise: rt = (Y*qw)*(Y*vw) / (Y*kw + eps); weights uniform per column
    const float qre = qr[kf], qim = qi[kf];
    const float kre = kr[kf], kim = ki[kf];
    const float vre = vr[kf], vim = vi[kf];
    const float rre = rr[kf], rim = ri[kf];
    for (int i = tid; i < NSEQ; i += 256) {
        float yr = re[i], yi = im[i];
        float aqr = yr * qre - yi * qim, aqi = yr * qim + yi * qre;
        float akr = yr * kre - yi * kim + 1e-12f, aki = yr * kim + yi * kre;
        float avr = yr * vre - yi * vim, avi = yr * vim + yi * vre;
        float tr = aqr * avr - aqi * avi;
        float ti = aqr * avi + aqi * avr;
        float inv = 1.0f / (akr * akr + aki * aki);
        re[i] = (tr * akr + ti * aki) * inv;
        im[i] = (ti * akr - tr * aki) * inv;
    }
    __syncthreads();

    // inverse FFT (sign = +1), DIF: natural in, bit-reversed out
    for (int len = NSEQ / 2; len >= 1; len >>= 1) {
        const float ang0 = 3.14159265358979323846f / (float)len;
        for (int t = tid; t < NSEQ / 2; t += 256) {
            int pos = t & (len - 1);
            int i0  = ((t & ~(len - 1)) << 1) | pos;
            int i1  = i0 + len;
            float ur = re[i0], ui = im[i0];
            float xr = re[i1], xi = im[i1];
            float c, s; __sincosf(ang0 * (float)pos, &s, &c);
            float dr = ur - xr, di = ui - xi;
            re[i0] = ur + xr;           im[i0] = ui + xi;
            re[i1] = dr * c - di * s;   im[i1] = dr * s + di * c;
        }
        __syncthreads();
    }

    // undo bit reversal on store; scale 1/N and multiply by rw
    const float scale = 1.0f / (float)NSEQ;
    for (int i = tid; i < NSEQ; i += 256) {
        int j = (int)(__brev((unsigned)i) >> (32 - LOG2_NSEQ));
        float ar = re[j] * scale, ai = im[j] * scale;
        wre[base + i] = ar * rre - ai * rim;
        wim[base + i] = ar * rim + ai * rre;
    }
}

// ---------------------------------------------------------------------------
// Kernel 3: inverse 128-pt DFT over size dim via f32 WMMA; real output.
//   out[b][s][n] = (1/128) * sum_k (Re*cos - Im*sin) = (Re*C + Im*(-S))/128
// ---------------------------------------------------------------------------
__global__ __launch_bounds__(256)
void k_idft_size(const float* __restrict__ wre, const float* __restrict__ wim,
                 const float* __restrict__ fcos, const float* __restrict__ fsin,
                 float* __restrict__ out) {
    __shared__ float reT[16 * 130];   // [s][k], padded
    __shared__ float imT[16 * 130];

    const int tid  = threadIdx.x;
    const int row0 = blockIdx.x * 16;     // (b,s) rows
    const int b    = row0 >> 13;
    const int s0   = row0 & (NSEQ - 1);

    // coalesced load [k][16 s], transpose into LDS [s][k]
    for (int i = tid; i < 128 * 16; i += 256) {
        int k = i >> 4, s = i & 15;
        size_t off = ((size_t)(b * NSZ + k)) * NSEQ + s0 + s;
        reT[s * 130 + k] = wre[off];
        imT[s * 130 + k] = wim[off];
    }
    __syncthreads();

    const int wave = tid >> 5, lane = tid & 31;
    const int hi = lane >> 4, lm = lane & 15;
    const int n = wave * 16 + lm;         // output size index

    v8f acc = {};
    for (int k4 = 0; k4 < 128; k4 += 4) {
        const int kb = k4 + 2 * hi;
        v2f aR; aR.x = reT[lm * 130 + kb];  aR.y = reT[lm * 130 + kb + 1];
        v2f aI; aI.x = imT[lm * 130 + kb];  aI.y = imT[lm * 130 + kb + 1];
        v2f bc; bc.x = fcos[n * 128 + kb];  bc.y = fcos[n * 128 + kb + 1];
        v2f bs; bs.x = fsin[n * 128 + kb];  bs.y = fsin[n * 128 + kb + 1];  // -sin
        acc = __builtin_amdgcn_wmma_f32_16x16x4_f32(false, aR, false, bc, (short)0, acc, false, false);
        acc = __builtin_amdgcn_wmma_f32_16x16x4_f32(false, aI, false, bs, (short)0, acc, false, false);
    }

    const float scale = 1.0f / 128.0f;
    for (int r = 0; r < 8; ++r) {
        int m = r + 8 * hi;               // local s
        out[((size_t)(row0 + m)) * NSZ + n] = acc[r] * scale;  // 64B coalesced in n
    }
}

// ---------------------------------------------------------------------------
extern "C" void kernel_launch(void* const* d_in, const int* in_sizes, int n_in,
                              void* d_out, int out_size, void* d_ws, size_t ws_size,
                              hipStream_t stream) {
    const float* x  = (const float*)d_in[0];
    const float* qr = (const float*)d_in[1];
    const float* qi = (const float*)d_in[2];
    const float* kr = (const float*)d_in[3];
    const float* ki = (const float*)d_in[4];
    const float* vr = (const float*)d_in[5];
    const float* vi = (const float*)d_in[6];
    const float* rr = (const float*)d_in[7];
    const float* ri = (const float*)d_in[8];
    float* out = (float*)d_out;

    float* ws   = (float*)d_ws;
    float* fcos = ws;
    float* fsin = ws + 128 * 128;
    float* wre  = ws + 2 * 128 * 128;
    float* wim  = wre + (size_t)NB * NSZ * NSEQ;

    k_init_dft    <<<64,   256, 0, stream>>>(fcos, fsin);
    k_dft_size_fwd<<<8192, 256, 0, stream>>>(x, fcos, fsin, wre, wim);
    k_seq_fft     <<<NB * NSZ, 256, 2 * NSEQ * sizeof(float), stream>>>(
        wre, wim, qr, qi, kr, ki, vr, vi, rr, ri);
    k_idft_size   <<<8192, 256, 0, stream>>>(wre, wim, fcos, fsin, out);
}